// GCN_27934467293291
// MI455X (gfx1250) — compile-verified
//
#include <hip/hip_runtime.h>
#include <math.h>

typedef __attribute__((ext_vector_type(2))) float v2f;
typedef __attribute__((ext_vector_type(8))) float v8f;

#define IN_CH  256
#define HID_CH 64
#define OUT_CH 16

// ---------------------------------------------------------------- init
__global__ void k_init(float* __restrict__ deg, float* __restrict__ agg1,
                       float* __restrict__ agg2, int N) {
  long long id = (long long)blockIdx.x * blockDim.x + threadIdx.x;
  long long totH = (long long)N * HID_CH;
  if (id < totH) agg1[id] = 0.0f;
  if (id < (long long)N * OUT_CH) agg2[id] = 0.0f;
  if (id < N) deg[id] = 1.0f;   // self-loop contributes 1 to every degree
}

// ------------------------------------------------------------- degrees
__global__ void k_degree(const long long* __restrict__ ei,
                         float* __restrict__ deg, long long E) {
  long long e = (long long)blockIdx.x * blockDim.x + threadIdx.x;
  if (e < E) {
    long long d = ei[E + e];             // dst row of edge_index
    atomicAdd(&deg[d], 1.0f);
  }
}

__global__ void k_rsqrt(float* __restrict__ dinv, int N) {
  int i = blockIdx.x * blockDim.x + threadIdx.x;
  if (i < N) dinv[i] = rsqrtf(dinv[i]); // deg >= 1 always (self-loop)
}

// ------------------------------------------- GEMM1: H1 = X @ W1  (f32 WMMA)
// Block = 256 threads = 8 waves; each wave owns a 16-row x 64-col tile.
// LDS holds W1 K-pair-interleaved: sW[(k>>1)*128 + n*2 + (k&1)]
// so a WMMA B operand {W1[k0][n], W1[k0+1][n]} is ONE ds_load_b64.
__global__ __launch_bounds__(256)
void k_gemm1(const float* __restrict__ X, const float* __restrict__ W1,
             float* __restrict__ H1, int N) {
  __shared__ float sW[IN_CH * HID_CH];           // 64 KB of the 320 KB LDS
  for (int idx = threadIdx.x; idx < IN_CH * HID_CH; idx += 256) {
    const int k = idx / HID_CH;
    const int n = idx - k * HID_CH;
    sW[(k >> 1) * (2 * HID_CH) + n * 2 + (k & 1)] = W1[idx];
  }
  __syncthreads();

  const int lane = threadIdx.x & 31;
  const int half = lane >> 4;                    // 0: K pair {0,1}; 1: {2,3}
  const int m    = lane & 15;                    // row within tile (A), col (B/D)
  const int row_base = blockIdx.x * 128 + (threadIdx.x >> 5) * 16;
  const int rowA = min(row_base + m, N - 1);     // clamp so EXEC stays all-1s
  const float* xrow = X + (long long)rowA * IN_CH;

  v8f acc0 = {}, acc1 = {}, acc2 = {}, acc3 = {};
  for (int kk = 0; kk < IN_CH; kk += 4) {
    const int k0 = kk + 2 * half;
    const v2f a = *(const v2f*)(xrow + k0);      // 64-bit global load
    const float* bp = &sW[(k0 >> 1) * (2 * HID_CH) + m * 2];
    const v2f b0 = *(const v2f*)(bp + 0);        // each: one ds_load_b64
    const v2f b1 = *(const v2f*)(bp + 32);
    const v2f b2 = *(const v2f*)(bp + 64);
    const v2f b3 = *(const v2f*)(bp + 96);
    acc0 = __builtin_amdgcn_wmma_f32_16x16x4_f32(false, a, false, b0, (short)0, acc0, false, false);
    acc1 = __builtin_amdgcn_wmma_f32_16x16x4_f32(false, a, false, b1, (short)0, acc1, false, false);
    acc2 = __builtin_amdgcn_wmma_f32_16x16x4_f32(false, a, false, b2, (short)0, acc2, false, false);
    acc3 = __builtin_amdgcn_wmma_f32_16x16x4_f32(false, a, false, b3, (short)0, acc3, false, false);
  }
#pragma unroll
  for (int v = 0; v < 8; ++v) {
    const int row = row_base + v + 8 * half;     // D layout: M = v + 8*(lane/16)
    if (row < N) {
      float* orow = H1 + (long long)row * HID_CH + m;
      orow[0]  = acc0[v];
      orow[16] = acc1[v];
      orow[32] = acc2[v];
      orow[48] = acc3[v];
    }
  }
}

// ---------------------------------------- layer-1 scatter (64 channels/edge)
__global__ void k_prop1(const long long* __restrict__ ei,
                        const float* __restrict__ dinv,
                        const float* __restrict__ H1,
                        float* __restrict__ agg1, long long E) {
  long long gid = (long long)blockIdx.x * blockDim.x + threadIdx.x;
  long long e = gid >> 4;                        // 16 threads/edge, float4 each
  if (e >= E) return;
  const int c4 = (int)(gid & 15) * 4;
  const long long s = ei[e];
  const long long d = ei[E + e];
  const float w = dinv[s] * dinv[d];
  const float4 v = *(const float4*)(H1 + s * HID_CH + c4);
  float* ap = agg1 + d * HID_CH + c4;
  atomicAdd(ap + 0, v.x * w);                    // global_atomic_add_f32 (L2)
  atomicAdd(ap + 1, v.y * w);
  atomicAdd(ap + 2, v.z * w);
  atomicAdd(ap + 3, v.w * w);
}

// ------------------------- self-loop + bias + ReLU, in place into H1 buffer
__global__ void k_actbias(float* __restrict__ H1, const float* __restrict__ agg1,
                          const float* __restrict__ dinv,
                          const float* __restrict__ b1, int N) {
  long long id = (long long)blockIdx.x * blockDim.x + threadIdx.x;
  if (id >= (long long)N * HID_CH) return;
  const int i = (int)(id >> 6);
  const int c = (int)(id & 63);
  const float di = dinv[i];
  const float v = agg1[id] + H1[id] * di * di + b1[c];
  H1[id] = fmaxf(v, 0.0f);
}

// ------------------------------------------- GEMM2: H2 = act @ W2 (f32 WMMA)
__global__ __launch_bounds__(256)
void k_gemm2(const float* __restrict__ A, const float* __restrict__ W2,
             float* __restrict__ H2, int N) {
  __shared__ float sW[HID_CH * OUT_CH];          // 4 KB, K-pair interleaved
  for (int idx = threadIdx.x; idx < HID_CH * OUT_CH; idx += 256) {
    const int k = idx / OUT_CH;
    const int n = idx - k * OUT_CH;
    sW[(k >> 1) * (2 * OUT_CH) + n * 2 + (k & 1)] = W2[idx];
  }
  __syncthreads();

  const int lane = threadIdx.x & 31;
  const int half = lane >> 4;
  const int m    = lane & 15;
  const int row_base = blockIdx.x * 128 + (threadIdx.x >> 5) * 16;
  const int rowA = min(row_base + m, N - 1);
  const float* arow = A + (long long)rowA * HID_CH;

  v8f acc = {};
  for (int kk = 0; kk < HID_CH; kk += 4) {
    const int k0 = kk + 2 * half;
    const v2f a = *(const v2f*)(arow + k0);
    const v2f b = *(const v2f*)(&sW[(k0 >> 1) * (2 * OUT_CH) + m * 2]);
    acc = __builtin_amdgcn_wmma_f32_16x16x4_f32(false, a, false, b, (short)0, acc, false, false);
  }
#pragma unroll
  for (int v = 0; v < 8; ++v) {
    const int row = row_base + v + 8 * half;
    if (row < N) H2[(long long)row * OUT_CH + m] = acc[v];
  }
}

// ---------------------------------------- layer-2 scatter (16 channels/edge)
__global__ void k_prop2(const long long* __restrict__ ei,
                        const float* __restrict__ dinv,
                        const float* __restrict__ H2,
                        float* __restrict__ agg2, long long E) {
  long long gid = (long long)blockIdx.x * blockDim.x + threadIdx.x;
  long long e = gid >> 2;                        // 4 threads/edge, float4 each
  if (e >= E) return;
  const int c4 = (int)(gid & 3) * 4;
  const long long s = ei[e];
  const long long d = ei[E + e];
  const float w = dinv[s] * dinv[d];
  const float4 v = *(const float4*)(H2 + s * OUT_CH + c4);
  float* ap = agg2 + d * OUT_CH + c4;
  atomicAdd(ap + 0, v.x * w);
  atomicAdd(ap + 1, v.y * w);
  atomicAdd(ap + 2, v.z * w);
  atomicAdd(ap + 3, v.w * w);
}

// --------------------- self-loop + bias + log_softmax over 16 classes
__global__ void k_final(const float* __restrict__ agg2, const float* __restrict__ H2,
                        const float* __restrict__ dinv, const float* __restrict__ b2,
                        float* __restrict__ out, int N) {
  int i = blockIdx.x * blockDim.x + threadIdx.x;
  if (i >= N) return;
  const float d2 = dinv[i] * dinv[i];
  const float* a = agg2 + (long long)i * OUT_CH;
  const float* h = H2   + (long long)i * OUT_CH;
  float v[OUT_CH];
  float mx = -INFINITY;
#pragma unroll
  for (int c = 0; c < OUT_CH; ++c) {
    v[c] = a[c] + h[c] * d2 + b2[c];
    mx = fmaxf(mx, v[c]);
  }
  float s = 0.0f;
#pragma unroll
  for (int c = 0; c < OUT_CH; ++c) s += expf(v[c] - mx);
  const float lse = mx + logf(s);
  float* o = out + (long long)i * OUT_CH;
#pragma unroll
  for (int c = 0; c < OUT_CH; ++c) o[c] = v[c] - lse;
}

// ----------------------------------------------------------------------------
extern "C" void kernel_launch(void* const* d_in, const int* in_sizes, int n_in,
                              void* d_out, int out_size, void* d_ws, size_t ws_size,
                              hipStream_t stream) {
  const float*     x  = (const float*)d_in[0];
  const long long* ei = (const long long*)d_in[1];   // edge_index int64 [2,E]
  const float*     W1 = (const float*)d_in[2];
  const float*     b1 = (const float*)d_in[3];
  const float*     W2 = (const float*)d_in[4];
  const float*     b2 = (const float*)d_in[5];
  float*           out = (float*)d_out;

  const int       N = in_sizes[0] / IN_CH;           // 100000
  const long long E = (long long)in_sizes[1] / 2;    // 1600000

  // workspace layout (floats): dinv | H1 | agg1 | H2 | agg2  (~64.5 MB total)
  float* ws = (float*)d_ws;
  size_t off = 0;
  float* dinv = ws + off; off += ((size_t)N + 255) & ~(size_t)255;
  float* H1   = ws + off; off += (size_t)N * HID_CH;
  float* agg1 = ws + off; off += (size_t)N * HID_CH;
  float* H2   = ws + off; off += (size_t)N * OUT_CH;
  float* agg2 = ws + off;

  const dim3 blk(256);
  const long long totH = (long long)N * HID_CH;

  k_init   <<<dim3((unsigned)((totH + 255) / 256)),   blk, 0, stream>>>(dinv, agg1, agg2, N);
  k_degree <<<dim3((unsigned)((E + 255) / 256)),      blk, 0, stream>>>(ei, dinv, E);
  k_rsqrt  <<<dim3((unsigned)((N + 255) / 256)),      blk, 0, stream>>>(dinv, N);
  k_gemm1  <<<dim3((unsigned)((N + 127) / 128)),      blk, 0, stream>>>(x, W1, H1, N);
  k_prop1  <<<dim3((unsigned)((E * 16 + 255) / 256)), blk, 0, stream>>>(ei, dinv, H1, agg1, E);
  k_actbias<<<dim3((unsigned)((totH + 255) / 256)),   blk, 0, stream>>>(H1, agg1, dinv, b1, N);
  k_gemm2  <<<dim3((unsigned)((N + 127) / 128)),      blk, 0, stream>>>(H1, W2, H2, N);
  k_prop2  <<<dim3((unsigned)((E * 4 + 255) / 256)),  blk, 0, stream>>>(ei, dinv, H2, agg2, E);
  k_final  <<<dim3((unsigned)((N + 255) / 256)),      blk, 0, stream>>>(agg2, H2, dinv, b2, out, N);
}